// RelativeBucketedTimeAndPositionBasedBias_33139967656423
// MI455X (gfx1250) — compile-verified
//
#include <hip/hip_runtime.h>
#include <cstddef>

// HSTU relative bucketed time + position bias for MI455X (gfx1250, wave32).
// out[b,i,j] = pos_w[N-1 + j - i] + ts_w[bucket], where
//   td     = (j <= i) ? ts[b, min(i+1,N-1)] - ts[b, j] : 0
//   bucket = clip(trunc(log(max(|td|,1)) / 0.301), 0, num_buckets)
//
// Pure streaming-store workload (256 MB out, ~11 us HBM floor @ 23.3 TB/s).
// CDNA5 features used: async global->LDS staging of ts_w (ASYNCcnt +
// s_wait_asynccnt), non-temporal b128 stores for the write stream.

typedef float v4f __attribute__((ext_vector_type(4)));
typedef int   v4i __attribute__((ext_vector_type(4)));

#define HSTU_LOG_BUCKET_DIV 0.301f

__global__ __launch_bounds__(512) void hstu_rel_bias_kernel(
    const int* __restrict__ ts,       // [B, N] sorted timestamps (int32)
    const float* __restrict__ ts_w,   // [num_buckets + 1]
    const float* __restrict__ pos_w,  // [2N - 1]
    float* __restrict__ out,          // [B, N, N]
    int N, int num_buckets)
{
    extern __shared__ float s_tsw[];  // num_buckets + 1 floats

    const int row = blockIdx.x;       // flat index over B*N rows
    const int b   = row / N;
    const int i   = row - b * N;

    // ---- Stage ts_w into LDS (gfx1250 async global->LDS path if available) ----
    const int nw = num_buckets + 1;
#if defined(__gfx1250__) && __has_builtin(__builtin_amdgcn_global_load_async_to_lds_b32)
    if ((int)threadIdx.x < nw) {
        __builtin_amdgcn_global_load_async_to_lds_b32(
            (__attribute__((address_space(1))) int*)(ts_w + threadIdx.x),
            (__attribute__((address_space(3))) int*)(s_tsw + threadIdx.x),
            /*imm offset*/ 0, /*cpol*/ 0);
    }
#if __has_builtin(__builtin_amdgcn_s_wait_asynccnt)
    __builtin_amdgcn_s_wait_asynccnt(0);
#else
    asm volatile("s_wait_asynccnt 0" ::: "memory");
#endif
#else
    for (int t = threadIdx.x; t < nw; t += blockDim.x) s_tsw[t] = ts_w[t];
#endif
    __syncthreads();

    // Uniform across the block -> scalar load.
    const int ii = (i + 1 < N) ? (i + 1) : (N - 1);
    const int ti = ts[(size_t)b * N + ii];

    const int*   __restrict__ tsrow = ts + (size_t)b * N;
    const float* __restrict__ pw    = pos_w + (N - 1 - i);
    float*       __restrict__ orow  = out + (size_t)row * N;

    // 512 threads x 4 elems covers N=2048 in one pass; loop kept for generality.
    for (int j0 = (int)threadIdx.x * 4; j0 < N; j0 += (int)blockDim.x * 4) {
        const v4i tj = *(const v4i*)(tsrow + j0);   // aligned b128 load
        v4f r;
#pragma unroll
        for (int k = 0; k < 4; ++k) {
            const int j = j0 + k;
            // Above the diagonal the reference zeroes td before max(|.|,1).
            float td = (j <= i) ? (float)(ti - tj[k]) : 0.0f;
            td = fmaxf(fabsf(td), 1.0f);
            int bkt = (int)(logf(td) / HSTU_LOG_BUCKET_DIV);  // log >= 0 -> trunc == clip-low
            bkt = bkt > num_buckets ? num_buckets : bkt;
            r[k] = pw[j] + s_tsw[bkt];
        }
        // Streaming 256 MB output: non-temporal 128-bit store.
        __builtin_nontemporal_store(r, (v4f*)(orow + j0));
    }
}

extern "C" void kernel_launch(void* const* d_in, const int* in_sizes, int n_in,
                              void* d_out, int out_size, void* d_ws, size_t ws_size,
                              hipStream_t stream) {
    const int*   ts    = (const int*)d_in[0];    // [B, N]
    const float* ts_w  = (const float*)d_in[1];  // [num_buckets + 1]
    const float* pos_w = (const float*)d_in[2];  // [2N - 1]
    float* out = (float*)d_out;

    const int N           = (in_sizes[2] + 1) / 2;  // pos_w has 2N-1 entries
    const int B           = in_sizes[0] / N;
    const int num_buckets = in_sizes[1] - 1;

    const dim3 grid((unsigned)(B * N));
    const dim3 block(512);
    const size_t lds_bytes = (size_t)(num_buckets + 1) * sizeof(float);

    hipLaunchKernelGGL(hstu_rel_bias_kernel, grid, block, lds_bytes, stream,
                       ts, ts_w, pos_w, out, N, num_buckets);
}